// EdgeAttention_369367188027
// MI455X (gfx1250) — compile-verified
//
#include <hip/hip_runtime.h>

// EdgeAttention: B=64, N=512, D=128, T=8
// out[b,i,j] = leakyrelu( mask(edges) * ( s_src[b,i,k] + s_dst[b,j,k] ) ),
//   k = clip(edges[b,i,j], 0, T-1),
//   s_src = src @ a_src^T, s_dst = dst @ a_dst^T   (a split from (2*D*T,1))

#define B_ 64
#define N_ 512
#define D_ 128
#define T_ 8
#define NEG_SLOPE 0.2f

typedef __attribute__((ext_vector_type(2))) float v2f;
typedef __attribute__((ext_vector_type(8))) float v8f;

// ---------------------------------------------------------------------------
// Kernel 1: projection GEMMs via V_WMMA_F32_16X16X4_F32.
// One wave (32 lanes) computes one 16-row tile of BOTH s_src and s_dst.
// A-frag (16x4 f32, 2 VGPRs/lane):  M = lane&15, K = step*4 + (lane<16?{0,1}:{2,3})
// B-frag (4x16 f32, 2 VGPRs/lane):  N(t) = lane&15, same K striping; cols t>=8 are 0.
// C/D (16x16 f32, 8 VGPRs/lane):    row M = v + (lane<16?0:8), col = lane&15.
// ---------------------------------------------------------------------------
__global__ __launch_bounds__(256)
void edgeattn_project(const float* __restrict__ src,
                      const float* __restrict__ dst,
                      const float* __restrict__ a,
                      float* __restrict__ s_src,
                      float* __restrict__ s_dst) {
    const int lane  = threadIdx.x & 31;
    const int tile  = blockIdx.x * 8 + (threadIdx.x >> 5);   // 16-row tile index
    const int mrow  = (tile << 4) + (lane & 15);             // row of A fragment
    const int khalf = (lane & 16) ? 2 : 0;                   // K sub-offset per step
    const int tcol  = lane & 15;                             // output column (t)

    // B fragment source: a[t*2D + d] (src half), a[t*2D + D + d] (dst half).
    // Lane-masked (no divergent branch) so EXEC is all-1s at every WMMA.
    const int   tt  = tcol & (T_ - 1);
    const float msk = (tcol < T_) ? 1.0f : 0.0f;
    const float* __restrict__ arow   = a + (size_t)tt * (2 * D_);
    const float* __restrict__ srcRow = src + (size_t)mrow * D_;
    const float* __restrict__ dstRow = dst + (size_t)mrow * D_;

    v8f c_src = {};
    v8f c_dst = {};

#pragma unroll 4
    for (int kk = 0; kk < D_ / 4; ++kk) {
        const int k0 = (kk << 2) + khalf;

        v2f a_s, a_d, b_s, b_d;
        a_s.x = srcRow[k0];     a_s.y = srcRow[k0 + 1];
        a_d.x = dstRow[k0];     a_d.y = dstRow[k0 + 1];
        b_s.x = arow[k0]        * msk;
        b_s.y = arow[k0 + 1]    * msk;
        b_d.x = arow[D_ + k0]   * msk;
        b_d.y = arow[D_ + k0 + 1] * msk;

        // D = A*B + C   (8 args: neg_a, A, neg_b, B, c_mod, C, reuse_a, reuse_b)
        c_src = __builtin_amdgcn_wmma_f32_16x16x4_f32(
            false, a_s, false, b_s, (short)0, c_src, false, false);
        c_dst = __builtin_amdgcn_wmma_f32_16x16x4_f32(
            false, a_d, false, b_d, (short)0, c_dst, false, false);
    }

    // Scatter the valid T=8 columns of the 16x16 accumulators.
    if (tcol < T_) {
        const int rbase = (tile << 4) + ((lane & 16) ? 8 : 0);
#pragma unroll
        for (int v = 0; v < 8; ++v) {
            s_src[(size_t)(rbase + v) * T_ + tcol] = c_src[v];
            s_dst[(size_t)(rbase + v) * T_ + tcol] = c_dst[v];
        }
    }
}

// ---------------------------------------------------------------------------
// Kernel 2: memory-bound edge pass. One thread = 4 consecutive j.
// Traffic: 67 MB edges in + 67 MB out; score tables (2 MB) stay hot in cache.
// ---------------------------------------------------------------------------
__global__ __launch_bounds__(256)
void edgeattn_score(const int* __restrict__ edges,
                    const float* __restrict__ s_src,
                    const float* __restrict__ s_dst,
                    float* __restrict__ out) {
    const size_t e   = ((size_t)blockIdx.x * 256 + threadIdx.x) * 4;
    const int    b   = (int)(e >> 18);            // / (N*N) = 262144
    const int    rem = (int)(e & (N_ * N_ - 1));
    const int    i   = rem >> 9;                  // / N
    const int    j   = rem & (N_ - 1);

    // Stream-ahead prefetch of the edge tensor (global_prefetch_b8).
    __builtin_prefetch((const void*)(edges + e + 8192), 0, 1);

    const int4 ev = *(const int4*)(edges + e);

    const float* __restrict__ sr = s_src + ((size_t)b * N_ + i) * T_;
    const float* __restrict__ sd = s_dst + ((size_t)b * N_ + j) * T_;

    float4 r;
    {
        int k = ev.x < 0 ? 0 : (ev.x > T_ - 1 ? T_ - 1 : ev.x);
        float g = sr[k] + sd[0 * T_ + k];
        g = (ev.x < 0) ? 0.0f : g;
        r.x = (g >= 0.0f) ? g : NEG_SLOPE * g;
    }
    {
        int k = ev.y < 0 ? 0 : (ev.y > T_ - 1 ? T_ - 1 : ev.y);
        float g = sr[k] + sd[1 * T_ + k];
        g = (ev.y < 0) ? 0.0f : g;
        r.y = (g >= 0.0f) ? g : NEG_SLOPE * g;
    }
    {
        int k = ev.z < 0 ? 0 : (ev.z > T_ - 1 ? T_ - 1 : ev.z);
        float g = sr[k] + sd[2 * T_ + k];
        g = (ev.z < 0) ? 0.0f : g;
        r.z = (g >= 0.0f) ? g : NEG_SLOPE * g;
    }
    {
        int k = ev.w < 0 ? 0 : (ev.w > T_ - 1 ? T_ - 1 : ev.w);
        float g = sr[k] + sd[3 * T_ + k];
        g = (ev.w < 0) ? 0.0f : g;
        r.w = (g >= 0.0f) ? g : NEG_SLOPE * g;
    }

    *(float4*)(out + e) = r;
}

// ---------------------------------------------------------------------------
extern "C" void kernel_launch(void* const* d_in, const int* in_sizes, int n_in,
                              void* d_out, int out_size, void* d_ws, size_t ws_size,
                              hipStream_t stream) {
    const float* src   = (const float*)d_in[0];   // (B,N,D) f32
    const float* dst   = (const float*)d_in[1];   // (B,N,D) f32
    const int*   edges = (const int*)d_in[2];     // (B,N,N) i32
    const float* a     = (const float*)d_in[3];   // (2*D*T,1) f32
    float*       out   = (float*)d_out;           // (B,N,N) f32

    float* s_src = (float*)d_ws;                          // B*N*T floats = 1 MB
    float* s_dst = s_src + (size_t)B_ * N_ * T_;          // + 1 MB

    // Kernel 1: (B*N/16) = 2048 wave-tiles, 8 waves per 256-thread block.
    edgeattn_project<<<(B_ * N_ / 16) / 8, 256, 0, stream>>>(src, dst, a, s_src, s_dst);

    // Kernel 2: B*N*N / 4 elements-per-thread / 256 threads-per-block.
    const int blocks = (B_ * N_ * N_) / (4 * 256);
    edgeattn_score<<<blocks, 256, 0, stream>>>(edges, s_src, s_dst, out);
}